// Discounted_Cumsum_53120155517198
// MI455X (gfx1250) — compile-verified
//
#include <hip/hip_runtime.h>

typedef float v8f __attribute__((ext_vector_type(8)));
typedef float v2f __attribute__((ext_vector_type(2)));

#define B_  4
#define H_  16
#define S_  4096
#define D_  128
#define LCH 256            // timesteps per chunk
#define NCH (S_ / LCH)     // 16 chunks per (b,h)

// ---------------------------------------------------------------------------
// Pass A: per-chunk discounted sum with zero seed -> chunk-final carry only.
// Streams the full input through L2 (RT hints) so pass C re-reads hit L2.
// ---------------------------------------------------------------------------
__global__ void __launch_bounds__(128)
dc_pass_a(const float* __restrict__ x, const float* __restrict__ gamma,
          float* __restrict__ carry) {
  const int blk = blockIdx.x;        // bh * NCH + c
  const int bh  = blk / NCH;
  const int c   = blk % NCH;
  const int h   = bh % H_;
  const int d   = threadIdx.x;
  const float g = gamma[h];

  const float* xp = x + ((size_t)bh * S_ + (size_t)c * LCH) * D_ + d;
  float acc = 0.0f;
#pragma unroll 8
  for (int t = 0; t < LCH; ++t) {
    acc = fmaf(g, acc, xp[(size_t)t * D_]);   // y = x + g*y
  }
  carry[(size_t)blk * D_ + d] = acc;
}

// ---------------------------------------------------------------------------
// Pass B: scan the NCH chunk carries per column with decay G = gamma^LCH.
// Rewrites each slot in place with the carry-IN for that chunk (P[c-1]).
// ---------------------------------------------------------------------------
__global__ void __launch_bounds__(256)
dc_pass_b(const float* __restrict__ gamma, float* __restrict__ carry) {
  const int col = blockIdx.x * 256 + threadIdx.x;  // bh*D_ + d
  const int bh  = col / D_;
  const int d   = col % D_;
  const int h   = bh % H_;

  float G = gamma[h];
#pragma unroll
  for (int i = 0; i < 8; ++i) G *= G;              // gamma^256

  float P = 0.0f;
  float* cp = carry + (size_t)bh * NCH * D_ + d;
#pragma unroll
  for (int c = 0; c < NCH; ++c) {
    const float t = cp[(size_t)c * D_];
    cp[(size_t)c * D_] = P;                        // carry-in for chunk c
    P = fmaf(G, P, t);                             // P[c] = carry[c] + G*P[c-1]
  }
}

// ---------------------------------------------------------------------------
// Pass C: seeded per-chunk scan on the matrix pipe.
// Per wave: one 16-wide d-tile, 16-step t-tiles. Y = M*X + C via 4 chained
// V_WMMA_F32_16X16X4_F32 (K split 16 = 4x4). M[m][j] = g^(m-j) (lower tri),
// C[m][n] = g^(m+1)*carry[n]. Carry = row 15 of D, pulled via wave32 shuffle.
// Output uses non-temporal stores so input lines survive in L2.
// ---------------------------------------------------------------------------
__global__ void __launch_bounds__(256)
dc_pass_c_wmma(const float* __restrict__ x, const float* __restrict__ gamma,
               const float* __restrict__ carry_in, float* __restrict__ y) {
  const int blk  = blockIdx.x;       // bh * NCH + c
  const int bh   = blk / NCH;
  const int c    = blk % NCH;
  const int h    = bh % H_;
  const int wv   = threadIdx.x >> 5; // 8 waves -> 8 d-tiles of 16
  const int lane = threadIdx.x & 31;
  const int n    = lane & 15;        // column within tile (also A-row m)
  const int hi   = lane >> 4;        // half-wave select
  const int d0   = wv * 16;
  const float g  = gamma[h];

  // Per-lane constant fragments.
  // A (16x4 f32 layout): lanes 0-15 row m=lane {K0,K1}; lanes 16-31 row m {K2,K3}.
  // ac[k][q] = M[m][4k + 2*hi + q] = g^(m-j) if j<=m else 0.
  // gc[r] = g^(m+1) with m = r + 8*hi  (C/D row held in VGPR r of this lane).
  float ac[4][2];
  float gc[8];
#pragma unroll
  for (int k = 0; k < 4; ++k) { ac[k][0] = 0.0f; ac[k][1] = 0.0f; }
#pragma unroll
  for (int r = 0; r < 8; ++r) gc[r] = 0.0f;
  {
    const int m = n;
    float p = 1.0f;                                  // g^e
#pragma unroll
    for (int e = 0; e <= 16; ++e) {
#pragma unroll
      for (int k = 0; k < 4; ++k) {
        const int j0 = 4 * k + 2 * hi;
        if (j0     <= m && (m - j0)     == e) ac[k][0] = p;
        if (j0 + 1 <= m && (m - j0 - 1) == e) ac[k][1] = p;
      }
#pragma unroll
      for (int r = 0; r < 8; ++r)
        if ((r + 8 * hi + 1) == e) gc[r] = p;
      p *= g;
    }
  }

  const size_t colbase = ((size_t)bh * S_ + (size_t)c * LCH) * D_ + (size_t)(d0 + n);
  const float* xp = x + colbase;
  float*       yp = y + colbase;

  float cw = carry_in[((size_t)bh * NCH + c) * D_ + d0 + n];   // y just before chunk

#pragma unroll 1
  for (int t16 = 0; t16 < LCH / 16; ++t16) {
    const int tb = t16 * 16;

    // C[m][n] = g^(m+1) * carry[n] as initial accumulator
    v8f acc;
#pragma unroll
    for (int r = 0; r < 8; ++r) acc[r] = gc[r] * cw;

    // Y = sum_k A_k * B_k + C   (K = 16 split into 4 slices)
#pragma unroll
    for (int k = 0; k < 4; ++k) {
      const int t0 = tb + 4 * k + 2 * hi;
      v2f Bv;
      Bv[0] = __builtin_nontemporal_load(xp + (size_t)t0 * D_);       // X[t0][n]
      Bv[1] = __builtin_nontemporal_load(xp + (size_t)(t0 + 1) * D_); // X[t0+1][n]
      v2f Av; Av[0] = ac[k][0]; Av[1] = ac[k][1];
      acc = __builtin_amdgcn_wmma_f32_16x16x4_f32(
          /*neg_a=*/false, Av, /*neg_b=*/false, Bv,
          /*c_mod=*/(short)0, acc, /*reuse_a=*/false, /*reuse_b=*/false);
    }

    // Store D tile (row = tb + r + 8*hi, col = d0+n), NT so L2 keeps the input
#pragma unroll
    for (int r = 0; r < 8; ++r)
      __builtin_nontemporal_store(acc[r], yp + (size_t)(tb + r + 8 * hi) * D_);

    // Next tile's carry = Y[15][n] = VGPR7 of hi-half lane with same n
    cw = __shfl(acc[7], n + 16, 32);
  }
}

// ---------------------------------------------------------------------------
extern "C" void kernel_launch(void* const* d_in, const int* in_sizes, int n_in,
                              void* d_out, int out_size, void* d_ws, size_t ws_size,
                              hipStream_t stream) {
  const float* x     = (const float*)d_in[0];   // (B,H,S,D) fp32
  const float* gamma = (const float*)d_in[1];   // (H,) fp32
  float* y  = (float*)d_out;
  float* ws = (float*)d_ws;                     // B*H*NCH*D floats = 512 KB

  dc_pass_a<<<dim3(B_ * H_ * NCH), dim3(128), 0, stream>>>(x, gamma, ws);
  dc_pass_b<<<dim3((B_ * H_ * D_) / 256), dim3(256), 0, stream>>>(gamma, ws);
  dc_pass_c_wmma<<<dim3(B_ * H_ * NCH), dim3(256), 0, stream>>>(x, gamma, ws, y);
}